// SpecAug_65962107732533
// MI455X (gfx1250) — compile-verified
//
#include <hip/hip_runtime.h>
#include <stdint.h>

// ---------------------------------------------------------------------------
// SpecAugment for x: f32[7, 100000, 128].
// Memory-bound: ~717 MB of traffic -> ~31 us at 23.3 TB/s. No matrix math, so
// no WMMA; instead exercise the CDNA5 TDM (tensor_load_to_lds + tensorcnt)
// and NT cache hints (dataset is 2x the 192MB L2 -> streaming policy).
// ---------------------------------------------------------------------------

#define LOG_EPS_F (-36.04365338911715f)

typedef __attribute__((ext_vector_type(4))) float        v4f;
typedef __attribute__((ext_vector_type(4))) unsigned int u32x4;
typedef __attribute__((ext_vector_type(8))) int          i32x8;
typedef __attribute__((ext_vector_type(4))) int          i32x4;

#if defined(__has_builtin)
#if __has_builtin(__builtin_amdgcn_tensor_load_to_lds)
#define SPECAUG_TDM 1
#endif
#endif
#ifndef SPECAUG_TDM
#define SPECAUG_TDM 0
#endif

// ---- JAX Threefry-2x32 (exact) --------------------------------------------
__device__ __forceinline__ void tf2x32(uint32_t k0, uint32_t k1,
                                       uint32_t x0, uint32_t x1,
                                       uint32_t& y0, uint32_t& y1) {
  const uint32_t ks0 = k0, ks1 = k1, ks2 = k0 ^ k1 ^ 0x1BD11BDAu;
  const uint32_t ks[3] = {ks0, ks1, ks2};
  const int R[8] = {13, 15, 26, 6, 17, 29, 16, 24};
  x0 += ks0;
  x1 += ks1;
#pragma unroll
  for (int g = 0; g < 5; ++g) {
#pragma unroll
    for (int i = 0; i < 4; ++i) {
      const int r = R[(g & 1) * 4 + i];
      x0 += x1;
      x1 = (x1 << r) | (x1 >> (32 - r));
      x1 ^= x0;
    }
    x0 += ks[(g + 1) % 3];
    x1 += ks[(g + 2) % 3] + (uint32_t)(g + 1);
  }
  y0 = x0;
  y1 = x1;
}

// random_bits(key, 32, (n,)) element idx; n = 2*half. JAX pairs (i, i+half).
__device__ __forceinline__ uint32_t jax_bits(uint32_t k0, uint32_t k1,
                                             uint32_t idx, uint32_t half) {
  uint32_t y0, y1;
  if (idx < half) {
    tf2x32(k0, k1, idx, idx + half, y0, y1);
    return y0;
  }
  tf2x32(k0, k1, idx - half, idx, y0, y1);
  return y1;
}

__device__ __forceinline__ float u01(uint32_t bits) {
  return __uint_as_float((bits >> 9) | 0x3F800000u) - 1.0f;
}

__device__ __forceinline__ void set_range(uint32_t fm[4], int s, int e) {
#pragma unroll
  for (int w = 0; w < 4; ++w) {
    const int lo = w * 32, hi = lo + 32;
    const int a = s > lo ? s : lo;
    const int b = e < hi ? e : hi;
    if (a < b) {
      const uint32_t m = (b - a == 32) ? 0xFFFFFFFFu
                                       : (((1u << (b - a)) - 1u) << (a - lo));
      fm[w] |= m;
    }
  }
}

// ---- Kernel 1: per-(channel, segment) mask parameters ----------------------
// ws[idx*8 + {0:tstart, 1:tend, 2..5:freq bitmask}] for idx = c*1000+s,
// c in [0,4), s in [0,1000). ws[32000] = apply_aug flag.
__global__ __launch_bounds__(256) void specaug_params(uint32_t* __restrict__ ws) {
  const uint32_t idx = blockIdx.x * 256u + threadIdx.x;

  // root key(42) = (0, 42); kp, km = split(key)  -> pairs (0,2), (1,3)
  uint32_t a0, a1, b0, b1;
  tf2x32(0u, 42u, 0u, 2u, a0, a1);
  tf2x32(0u, 42u, 1u, 3u, b0, b1);
  const uint32_t kp0 = a0, kp1 = b0, km0 = a1, km1 = b1;

  // k1..k4 = split(km, 4) -> pairs (i, i+4), i = 0..3
  uint32_t s0y0, s0y1, s1y0, s1y1, s2y0, s2y1, s3y0, s3y1;
  tf2x32(km0, km1, 0u, 4u, s0y0, s0y1);
  tf2x32(km0, km1, 1u, 5u, s1y0, s1y1);
  tf2x32(km0, km1, 2u, 6u, s2y0, s2y1);
  tf2x32(km0, km1, 3u, 7u, s3y0, s3y1);
  const uint32_t k1a = s0y0, k1b = s1y0;   // tlen
  const uint32_t k2a = s2y0, k2b = s3y0;   // tstart uniform
  const uint32_t k3a = s0y1, k3b = s1y1;   // flen
  const uint32_t k4a = s2y1, k4b = s3y1;   // fstart uniform

  if (idx == 0) {
    uint32_t p0, p1;
    tf2x32(kp0, kp1, 0u, 0u, p0, p1);      // uniform(kp, ()) (odd-size pad)
    ws[32000] = (u01(p0) <= 0.5f) ? 1u : 0u;
  }
  if (idx >= 4000u) return;

  // randint(k1, (4,1000), 0, 35): bits % 35 == JAX's split-half modular form
  const uint32_t tlen = jax_bits(k1a, k1b, idx, 2000u) % 35u;
  const float ut = u01(jax_bits(k2a, k2b, idx, 2000u));
  const uint32_t tstart = (uint32_t)(int)(ut * (float)(100u - tlen));

  const uint32_t e0 = 2u * idx, e1 = 2u * idx + 1u;
  const uint32_t fl0 = jax_bits(k3a, k3b, e0, 4000u) % 30u;
  const uint32_t fl1 = jax_bits(k3a, k3b, e1, 4000u) % 30u;
  const float high = (float)(128u - (fl0 > fl1 ? fl0 : fl1));
  const int fs0 = (int)(u01(jax_bits(k4a, k4b, e0, 4000u)) * high);
  const int fs1 = (int)(u01(jax_bits(k4a, k4b, e1, 4000u)) * high);

  uint32_t fm[4] = {0u, 0u, 0u, 0u};
  set_range(fm, fs0, fs0 + (int)fl0);
  set_range(fm, fs1, fs1 + (int)fl1);

  uint32_t* p = ws + (size_t)idx * 8u;
  p[0] = tstart;
  p[1] = tstart + tlen;
  p[2] = fm[0]; p[3] = fm[1]; p[4] = fm[2]; p[5] = fm[3];
  p[6] = 0u;    p[7] = 0u;
}

// ---- Kernel 2: apply mask, one (c, seg) tile of 100x128 f32 per block ------
__global__ __launch_bounds__(320) void specaug_apply(const float* __restrict__ x,
                                                     float* __restrict__ out,
                                                     const uint32_t* __restrict__ params) {
  const int c = blockIdx.x / 1000;
  const int s = blockIdx.x - c * 1000;

  const uint32_t apply = params[32000];
  uint32_t tstart = 0u, tend = 0u, fw0 = 0u, fw1 = 0u, fw2 = 0u, fw3 = 0u;
  if (apply && c < 4) {
    const uint32_t* p = params + (size_t)(c * 1000 + s) * 8u;
    tstart = p[0]; tend = p[1];
    fw0 = p[2]; fw1 = p[3]; fw2 = p[4]; fw3 = p[5];
  }

  const size_t base  = ((size_t)c * 100000u + (size_t)s * 100u) * 128u; // floats
  const size_t base4 = base >> 2;                                      // float4s

#if SPECAUG_TDM
  __shared__ __align__(16) float tile[12800];   // 51200 B: full tile in LDS
  if (threadIdx.x < 32) {                       // wave-uniform: wave 0 only
    const uint64_t ga = (uint64_t)(uintptr_t)(x + base);
    u32x4 g0;
    g0.x = 1u;                                  // count=1, user descriptor
    g0.y = (uint32_t)(uintptr_t)(&tile[0]);     // lds_addr (low 32 bits)
    g0.z = (uint32_t)ga;                        // global_addr[31:0]
    g0.w = (uint32_t)(ga >> 32) | (2u << 30);   // global_addr[56:32] | type=2
    i32x8 g1;
    g1[0] = (int)(2u << 16);                    // data_size=4B, wg_mask=0
    g1[1] = (int)(12800u << 16);                // tensor_dim0[15:0]
    g1[2] = 0;                                  // tensor_dim0[31:16], dim1 lo
    g1[3] = (int)(12800u << 16);                // tile_dim0 = 12800 (1-D tile)
    g1[4] = 0;                                  // tile_dim1 = tile_dim2 = 0
    g1[5] = (int)12800u;                        // tensor_dim0_stride lo
    g1[6] = 0;
    g1[7] = 0;
    i32x4 gz4 = {0, 0, 0, 0};
    i32x8 gz8 = {0, 0, 0, 0, 0, 0, 0, 0};
    // 6-arg form (amdgpu-toolchain clang-23 / therock-10.0 headers)
    __builtin_amdgcn_tensor_load_to_lds(g0, g1, gz4, gz4, gz8, 0);
#if defined(__has_builtin) && __has_builtin(__builtin_amdgcn_s_wait_tensorcnt)
    __builtin_amdgcn_s_wait_tensorcnt(0);
#else
    asm volatile("s_wait_tensorcnt 0" ::: "memory");
#endif
  }
  __syncthreads();
#endif

  const int tid = (int)threadIdx.x;             // 320 = 10 waves
  const int f4  = tid & 31;                     // float4 column: loop-invariant
  const uint32_t wlo  = (f4 & 8)  ? fw1 : fw0;
  const uint32_t whi  = (f4 & 8)  ? fw3 : fw2;
  const uint32_t word = (f4 & 16) ? whi : wlo;
  const uint32_t nib  = (word >> ((f4 & 7) * 4)) & 0xFu;
  const int row0 = tid >> 5;                    // rows step by 10 per iter

  v4f* __restrict__ o4 = (v4f*)out + base4;
#if SPECAUG_TDM
  const v4f* __restrict__ t4 = (const v4f*)tile;
#else
  const v4f* __restrict__ t4 = (const v4f*)x + base4;
#endif

#pragma unroll
  for (int k = 0; k < 10; ++k) {
    const int i   = tid + k * 320;
    const int row = row0 + k * 10;
    v4f v;
#if SPECAUG_TDM
    v = t4[i];                                  // ds_load_b128
#else
    v = __builtin_nontemporal_load(t4 + i);     // global_load_b128 TH=NT
#endif
    const bool tb = ((uint32_t)row >= tstart) && ((uint32_t)row < tend);
    const uint32_t m = tb ? 0xFu : nib;
    v.x = (m & 1u) ? LOG_EPS_F : v.x;
    v.y = (m & 2u) ? LOG_EPS_F : v.y;
    v.z = (m & 4u) ? LOG_EPS_F : v.z;
    v.w = (m & 8u) ? LOG_EPS_F : v.w;
    __builtin_nontemporal_store(v, o4 + i);     // global_store_b128 TH=NT
  }
}

// ---------------------------------------------------------------------------
extern "C" void kernel_launch(void* const* d_in, const int* in_sizes, int n_in,
                              void* d_out, int out_size, void* d_ws, size_t ws_size,
                              hipStream_t stream) {
  (void)in_sizes; (void)n_in; (void)out_size; (void)ws_size;
  const float* x  = (const float*)d_in[0];
  float* out      = (float*)d_out;
  uint32_t* ws    = (uint32_t*)d_ws;            // 4000*8 u32 params + flag

  specaug_params<<<16, 256, 0, stream>>>(ws);   // 4096 threads >= 4000 + flag
  specaug_apply<<<7000, 320, 0, stream>>>(x, out, ws);
}